// VLLMFP8KVCache_7103875908080
// MI455X (gfx1250) — compile-verified
//
#include <hip/hip_runtime.h>
#include <hip/hip_bf16.h>
#include <stdint.h>

// FP8 KV-cache round trip for MI455X (gfx1250).
//   inv[slot] = token that wrote the slot (or -1)  -> built in d_ws (512 KB)
//   gather:  out[f] = bf16( e4m3fn( src_row ) ),  src_row = input[t] if hit
//                                                 else cache[slot]
// Memory-bound: ~400 MB traffic -> ~17 us at 23.3 TB/s.

#define HD 1024  // H * D = 8 * 128 (fixed by the reference)

// ---------------- fp8 e4m3fn quantize->dequantize round trip ----------------
#if __has_builtin(__builtin_amdgcn_cvt_pk_fp8_f32) && \
    __has_builtin(__builtin_amdgcn_cvt_pk_f32_fp8)
// Hardware path: CDNA5 fp8 is OCP E4M3 (not fnuz), RNE — matches
// jnp.float8_e4m3fn.  Two VALU ops per element pair.
typedef __attribute__((ext_vector_type(2))) float v2f_t;
__device__ __forceinline__ void qdq_pair(float x0, float x1, float& y0, float& y1) {
    int packed = __builtin_amdgcn_cvt_pk_fp8_f32(x0, x1, 0, /*word=*/false);
    v2f_t r = __builtin_amdgcn_cvt_pk_f32_fp8(packed, /*word=*/false);
    y0 = r[0];
    y1 = r[1];
}
#else
// Software fallback: RNE quantization to the e4m3fn grid.
__device__ __forceinline__ float qdq_one(float x) {
    unsigned u = __float_as_uint(x);
    unsigned s = u & 0x80000000u;
    unsigned a = u & 0x7fffffffu;
    if (a >= 0x7f800000u)                       // NaN / Inf -> NaN (e4m3fn has no Inf)
        return __uint_as_float(s | 0x7fc00000u);
    float ax = __uint_as_float(a);
    float r;
    if (ax < 0.015625f) {                       // |x| < 2^-6: subnormals, step 2^-9
        r = rintf(ax * 512.0f) * (1.0f / 512.0f);
    } else {                                    // RNE to 3 mantissa bits
        unsigned lsb = (a >> 20) & 1u;
        unsigned q = (a + 0x000FFFFFu + lsb) & 0xFFF00000u;
        r = __uint_as_float(q);
        if (r > 448.0f)                          // overflow -> NaN
            r = __uint_as_float(0x7fc00000u);
    }
    return __uint_as_float(__float_as_uint(r) | s);
}
__device__ __forceinline__ void qdq_pair(float x0, float x1, float& y0, float& y1) {
    y0 = qdq_one(x0);
    y1 = qdq_one(x1);
}
#endif

// Every e4m3 value is exactly representable in bf16, so truncation == RNE.
__device__ __forceinline__ unsigned pack_bf16(float lo, float hi) {
    return (__float_as_uint(lo) >> 16) | (__float_as_uint(hi) & 0xFFFF0000u);
}

// ---------------- kernel 1: inv[i] = -1 ----------------
__global__ void fp8kv_init_inv(int* __restrict__ inv, int n) {
    int i = blockIdx.x * blockDim.x + threadIdx.x;
    if (i < n) inv[i] = -1;
}

// ---------------- kernel 2: inv[slot_mapping[t]] = t ----------------
__global__ void fp8kv_build_inv(const int* __restrict__ slot_mapping,
                                int* __restrict__ inv, int T) {
    int t = blockIdx.x * blockDim.x + threadIdx.x;
    if (t < T) inv[slot_mapping[t]] = t;
}

// ---------------- kernel 3: gather + quant/dequant ----------------
// 256 threads = 2 rows/workgroup; 128 threads (4 wave32) per row, 8 elems each.
// The per-row hit/miss decision is uniform across each wave32.
__global__ __launch_bounds__(256) void fp8kv_gather(
    const float* __restrict__ input,   // [T, HD]
    const float* __restrict__ cache,   // [NUM_SLOTS, HD]
    const int*   __restrict__ blocks,  // [F]
    const int*   __restrict__ inv,     // [NUM_SLOTS]
    uint4*       __restrict__ out,     // [F, HD/8] packed bf16
    int F) {
    const int rid  = threadIdx.x >> 7;   // row within workgroup (0..1)
    const int lane = threadIdx.x & 127;  // 0..127, 8 contiguous elems each
    const int row  = blockIdx.x * 2 + rid;
    if (row >= F) return;

    const int slot = blocks[row];
    const int t    = inv[slot];
    const float* src = (t >= 0 ? input + (size_t)t * HD
                               : cache + (size_t)slot * HD) + lane * 8;

    float4 a0, a1;
#if defined(__gfx1250__) && \
    __has_builtin(__builtin_amdgcn_global_load_async_to_lds_b128)
    // CDNA5 async data-mover path: stage the row through LDS with
    // global_load_async_to_lds_b128, tracked by ASYNCcnt.
    __shared__ float stage[2][HD];
    {
        // Builtin signature (from hipcc diagnostic): pointee is a 16-byte int
        // vector; global source in addrspace(1), LDS dest in addrspace(3).
        typedef int v4i_t __attribute__((vector_size(16)));
        typedef v4i_t __attribute__((address_space(1)))* gv4i_p;
        typedef v4i_t __attribute__((address_space(3)))* lv4i_p;

        float* sp = &stage[rid][lane * 8];
        gv4i_p g0 = (gv4i_p)(float*)src;
        gv4i_p g1 = (gv4i_p)(float*)(src + 4);
        lv4i_p l0 = (lv4i_p)(void*)sp;
        lv4i_p l1 = (lv4i_p)(void*)(sp + 4);
        __builtin_amdgcn_global_load_async_to_lds_b128(g0, l0, 0, 0);
        __builtin_amdgcn_global_load_async_to_lds_b128(g1, l1, 0, 0);
#if __has_builtin(__builtin_amdgcn_s_wait_asynccnt)
        __builtin_amdgcn_s_wait_asynccnt(0);
#else
        asm volatile("s_wait_asynccnt 0" ::: "memory");
#endif
        a0 = *(const float4*)sp;
        a1 = *(const float4*)(sp + 4);
    }
#else
    a0 = *(const float4*)src;
    a1 = *(const float4*)(src + 4);
#endif

    float q[8];
    qdq_pair(a0.x, a0.y, q[0], q[1]);
    qdq_pair(a0.z, a0.w, q[2], q[3]);
    qdq_pair(a1.x, a1.y, q[4], q[5]);
    qdq_pair(a1.z, a1.w, q[6], q[7]);

    uint4 res;
    res.x = pack_bf16(q[0], q[1]);
    res.y = pack_bf16(q[2], q[3]);
    res.z = pack_bf16(q[4], q[5]);
    res.w = pack_bf16(q[6], q[7]);
    out[(size_t)row * (HD / 8) + lane] = res;  // one 128-bit store per lane
}

// ---------------- host launcher ----------------
extern "C" void kernel_launch(void* const* d_in, const int* in_sizes, int n_in,
                              void* d_out, int out_size, void* d_ws, size_t ws_size,
                              hipStream_t stream) {
    const float* input        = (const float*)d_in[0];  // [T, H, D] fp32
    const float* cache        = (const float*)d_in[1];  // [NUM_SLOTS, H, D] fp32
    const int*   slot_mapping = (const int*)d_in[2];    // [T]
    const int*   blocks       = (const int*)d_in[3];    // [F]

    const int T         = in_sizes[2];
    const int F         = in_sizes[3];
    const int num_slots = in_sizes[1] / HD;

    int* inv = (int*)d_ws;  // num_slots * 4 bytes (512 KB for the reference shapes)

    fp8kv_init_inv<<<(num_slots + 255) / 256, 256, 0, stream>>>(inv, num_slots);
    fp8kv_build_inv<<<(T + 255) / 256, 256, 0, stream>>>(slot_mapping, inv, T);
    fp8kv_gather<<<(F + 1) / 2, 256, 0, stream>>>(
        input, cache, blocks, inv, (uint4*)d_out, F);
}